// AlignedTriLoss_4475355922613
// MI455X (gfx1250) — compile-verified
//
#include <hip/hip_runtime.h>
#include <hip/hip_bf16.h>

typedef float v2f __attribute__((ext_vector_type(2)));
typedef float v8f __attribute__((ext_vector_type(8)));

#define N_SAMPLES 4096
#define DG        2048
#define M_LOC     8
#define C_LOC     256
#define MARGIN_F  0.3f
#define EPS_F     1e-12f

// Padded LDS row stride for the A strip: 2052 % 64 == 4 banks -> the 16 rows
// read by lanes 0..15 of a half-wave hit 16 distinct banks. Also a multiple
// of 4 so float4 stores stay aligned.
#define A_STRIDE (DG + 4)

// ---------------------------------------------------------------------------
// sq[row] = sum(gf[row]^2)
// ---------------------------------------------------------------------------
__global__ void k_rowsumsq(const float* __restrict__ gf, float* __restrict__ sq) {
    const int row = blockIdx.x;
    const float4* g = (const float4*)(gf + (size_t)row * DG);
    float s = 0.f;
    for (int j = threadIdx.x; j < DG / 4; j += blockDim.x) {
        float4 v = g[j];
        s += v.x * v.x + v.y * v.y + v.z * v.z + v.w * v.w;
    }
#pragma unroll
    for (int m = 16; m >= 1; m >>= 1) s += __shfl_xor(s, m, 32);
    __shared__ float wsum[8];
    const int lane = threadIdx.x & 31, wave = threadIdx.x >> 5;
    if (lane == 0) wsum[wave] = s;
    __syncthreads();
    if (threadIdx.x == 0) {
        float t = 0.f;
        for (int w = 0; w < 8; ++w) t += wsum[w];
        sq[row] = t;
    }
}

// ---------------------------------------------------------------------------
// Fused pairwise-distance GEMM (f32 WMMA 16x16x4) + hard positive/negative
// mining. One block = one 16-row strip of D = gf @ gf^T; A strip staged in
// LDS once; 8 waves sweep the 256 column tiles.
// ---------------------------------------------------------------------------
__global__ void __launch_bounds__(256)
k_pairwise_hard(const float* __restrict__ gf, const float* __restrict__ sq,
                const int* __restrict__ targets,
                float* __restrict__ dist_ap, int* __restrict__ ap_y,
                float* __restrict__ dist_an, int* __restrict__ an_y) {
    __shared__ float As[16 * A_STRIDE];
    __shared__ unsigned long long smax[8][16];
    __shared__ unsigned long long smin[8][16];

    const int tid  = threadIdx.x;
    const int lane = tid & 31;
    const int wave = tid >> 5;
    const int half = lane >> 4;   // which 16-lane half of the wave
    const int l16  = lane & 15;
    const int rb   = blockIdx.x * 16;

    // Stage the 16 x 2048 A strip (contiguous in global) into padded LDS.
    {
        const float4* g = (const float4*)(gf + (size_t)rb * DG);
        for (int idx = tid; idx < 16 * (DG / 4); idx += 256) {
            const int row = idx >> 9;        // DG/4 == 512
            const int q   = idx & 511;
            ((float4*)(As + row * A_STRIDE))[q] = g[row * (DG / 4) + q];
        }
    }
    __syncthreads();

    // Row metadata for the 8 C rows this lane holds (rows rb + r + 8*half).
    float sr[8];
    int   tr[8];
#pragma unroll
    for (int r = 0; r < 8; ++r) {
        const int row = rb + r + 8 * half;
        sr[r] = sq[row];
        tr[r] = targets[row];
    }

    // Packed (value<<32 | index) running keys. d >= sqrt(EPS) > 0 so the f32
    // bit pattern is monotone as an unsigned int. For argmax ties -> lowest
    // index we store ~col; for argmin ties -> lowest index we store col.
    unsigned long long kmax[8], kmin[8];
#pragma unroll
    for (int r = 0; r < 8; ++r) { kmax[r] = 0ull; kmin[r] = ~0ull; }

    const float* Arow = As + l16 * A_STRIDE + 2 * half;

    for (int ct = wave; ct < N_SAMPLES / 16; ct += 8) {
        const int cb = ct * 16;
        const float* Brow = gf + (size_t)(cb + l16) * DG + 2 * half;
        v8f c = {0.f, 0.f, 0.f, 0.f, 0.f, 0.f, 0.f, 0.f};
#pragma unroll 8
        for (int kb = 0; kb < DG; kb += 4) {
            v2f a = *(const v2f*)(Arow + kb);   // LDS b64 load
            v2f b = *(const v2f*)(Brow + kb);   // global b64 load (L2 resident)
            c = __builtin_amdgcn_wmma_f32_16x16x4_f32(
                    false, a, false, b, (short)0, c, false, false);
        }
        const int   col = cb + l16;
        const float sqc = sq[col];
        const int   tc  = targets[col];
#pragma unroll
        for (int r = 0; r < 8; ++r) {
            const float d2 = sr[r] + sqc - 2.0f * c[r];
            const float d  = sqrtf(fmaxf(d2, EPS_F));
            const unsigned fb = __float_as_uint(d);
            if (tr[r] == tc) {
                const unsigned long long k =
                    ((unsigned long long)fb << 32) | (unsigned)(~col);
                kmax[r] = (k > kmax[r]) ? k : kmax[r];
            } else {
                const unsigned long long k =
                    ((unsigned long long)fb << 32) | (unsigned)col;
                kmin[r] = (k < kmin[r]) ? k : kmin[r];
            }
        }
    }

    // Reduce across the 16 lanes of each half-wave (xor masks <= 8 stay
    // inside each 16-lane group).
#pragma unroll
    for (int m = 1; m <= 8; m <<= 1) {
#pragma unroll
        for (int r = 0; r < 8; ++r) {
            const unsigned long long om = __shfl_xor(kmax[r], m, 32);
            const unsigned long long on = __shfl_xor(kmin[r], m, 32);
            kmax[r] = (om > kmax[r]) ? om : kmax[r];
            kmin[r] = (on < kmin[r]) ? on : kmin[r];
        }
    }
    if (l16 == 0) {
#pragma unroll
        for (int r = 0; r < 8; ++r) {
            smax[wave][r + 8 * half] = kmax[r];
            smin[wave][r + 8 * half] = kmin[r];
        }
    }
    __syncthreads();
    if (tid < 16) {
        unsigned long long km = 0ull, kn = ~0ull;
        for (int w = 0; w < 8; ++w) {
            km = (smax[w][tid] > km) ? smax[w][tid] : km;
            kn = (smin[w][tid] < kn) ? smin[w][tid] : kn;
        }
        dist_ap[rb + tid] = __uint_as_float((unsigned)(km >> 32));
        ap_y[rb + tid]    = (int)(~(unsigned)km);
        dist_an[rb + tid] = __uint_as_float((unsigned)(kn >> 32));
        an_y[rb + tid]    = (int)(unsigned)kn;
    }
}

// ---------------------------------------------------------------------------
// global_loss partial sums
// ---------------------------------------------------------------------------
__global__ void k_global_loss(const float* __restrict__ dist_ap,
                              const float* __restrict__ dist_an,
                              float* __restrict__ accum) {
    const int i = blockIdx.x * blockDim.x + threadIdx.x;
    float t = fmaxf(dist_ap[i] - dist_an[i] + MARGIN_F, 0.f);
#pragma unroll
    for (int m = 16; m >= 1; m >>= 1) t += __shfl_xor(t, m, 32);
    __shared__ float wsum[8];
    const int lane = threadIdx.x & 31, wave = threadIdx.x >> 5;
    if (lane == 0) wsum[wave] = t;
    __syncthreads();
    if (threadIdx.x == 0) {
        float s = 0.f;
        for (int w = 0; w < 8; ++w) s += wsum[w];
        atomicAdd(accum, s);
    }
}

// ---------------------------------------------------------------------------
// 8x8 monotone shortest-path DP (matches _shortest_dist)
// ---------------------------------------------------------------------------
__device__ __forceinline__ float dp8x8(const float* D) {
    float prev[8];
    prev[0] = D[0];
#pragma unroll
    for (int j = 1; j < 8; ++j) prev[j] = prev[j - 1] + D[j];
#pragma unroll
    for (int i = 1; i < 8; ++i) {
        float c = prev[0] + D[i * 8];
        float nprev[8];
        nprev[0] = c;
#pragma unroll
        for (int j = 1; j < 8; ++j) {
            c = fminf(prev[j], c) + D[i * 8 + j];
            nprev[j] = c;
        }
#pragma unroll
        for (int j = 0; j < 8; ++j) prev[j] = nprev[j];
    }
    return prev[7];
}

// ---------------------------------------------------------------------------
// Local branch: one block per sample. tid<64 -> pos 8x8 entry, tid>=64 ->
// neg 8x8 entry. Each thread computes one (m,k) distance over C=256.
// ---------------------------------------------------------------------------
__global__ void k_local_loss(const float* __restrict__ lf,
                             const int* __restrict__ ap_y,
                             const int* __restrict__ an_y,
                             float* __restrict__ accum) {
    const int n   = blockIdx.x;
    const int tid = threadIdx.x;
    const int e   = tid & 63;
    const int m   = e >> 3;
    const int k   = e & 7;
    const int other = (tid < 64) ? ap_y[n] : an_y[n];

    const float4* x4 =
        (const float4*)(lf + (size_t)n * (M_LOC * C_LOC) + m * C_LOC);
    const float4* y4 =
        (const float4*)(lf + (size_t)other * (M_LOC * C_LOC) + k * C_LOC);

    float sx = 0.f, sy = 0.f, dt = 0.f;
#pragma unroll 4
    for (int c4 = 0; c4 < C_LOC / 4; ++c4) {
        const float4 a = x4[c4], b = y4[c4];
        sx += a.x * a.x + a.y * a.y + a.z * a.z + a.w * a.w;
        sy += b.x * b.x + b.y * b.y + b.z * b.z + b.w * b.w;
        dt += a.x * b.x + a.y * b.y + a.z * b.z + a.w * b.w;
    }
    const float d2 = sx + sy - 2.f * dt;
    const float d  = sqrtf(fmaxf(d2, EPS_F));
    const float ed = expf(d);
    const float t  = (ed - 1.f) / (ed + 1.f);

    __shared__ float Dp[64], Dn[64];
    __shared__ float res[2];
    if (tid < 64) Dp[e] = t; else Dn[e] = t;
    __syncthreads();
    if (tid == 0)  res[0] = dp8x8(Dp);
    if (tid == 64) res[1] = dp8x8(Dn);
    __syncthreads();
    if (tid == 0) {
        const float term = fmaxf(res[0] - res[1] + MARGIN_F, 0.f);
        atomicAdd(accum + 1, term);
    }
}

__global__ void k_init(float* accum) { accum[0] = 0.f; accum[1] = 0.f; }

__global__ void k_finalize(const float* __restrict__ accum,
                           float* __restrict__ out) {
    out[0] = accum[0] * (1.0f / (float)N_SAMPLES);
    out[1] = accum[1] * (1.0f / (float)N_SAMPLES);
}

// ---------------------------------------------------------------------------
extern "C" void kernel_launch(void* const* d_in, const int* in_sizes, int n_in,
                              void* d_out, int out_size, void* d_ws, size_t ws_size,
                              hipStream_t stream) {
    const float* gf      = (const float*)d_in[0];
    const float* lf      = (const float*)d_in[1];
    const int*   targets = (const int*)d_in[2];

    float* sq      = (float*)d_ws;
    float* dist_ap = sq + N_SAMPLES;
    int*   ap_y    = (int*)(dist_ap + N_SAMPLES);
    float* dist_an = (float*)(ap_y + N_SAMPLES);
    int*   an_y    = (int*)(dist_an + N_SAMPLES);
    float* accum   = (float*)(an_y + N_SAMPLES);   // 2 floats
    float* out     = (float*)d_out;

    k_init<<<1, 1, 0, stream>>>(accum);
    k_rowsumsq<<<N_SAMPLES, 256, 0, stream>>>(gf, sq);
    k_pairwise_hard<<<N_SAMPLES / 16, 256, 0, stream>>>(gf, sq, targets,
                                                        dist_ap, ap_y,
                                                        dist_an, an_y);
    k_global_loss<<<16, 256, 0, stream>>>(dist_ap, dist_an, accum);
    k_local_loss<<<N_SAMPLES, 128, 0, stream>>>(lf, ap_y, an_y, accum);
    k_finalize<<<1, 1, 0, stream>>>(accum, out);

    (void)in_sizes; (void)n_in; (void)out_size; (void)ws_size;
}